// KD_loss_13340168421463
// MI455X (gfx1250) — compile-verified
//
#include <hip/hip_runtime.h>
#include <math.h>

// ---------------------------------------------------------------------------
// KD loss: segmented class-mean pooling of two logits tensors + symmetric KL.
//   s_sum[k+5b, c] = sum_{vox: gt==k} src[b,c,vox]   via V_WMMA_F32_16X16X4_F32
//   (A = one-hot from gt, B = 4x16 channel slab, B col 5 == 1.0 -> nvox free)
// Memory-bound: ~216 MB read once -> ~10 us at 23.3 TB/s.
// ---------------------------------------------------------------------------

typedef float v2f __attribute__((ext_vector_type(2)));
typedef float v8f __attribute__((ext_vector_type(8)));

#define BATCH 2
#define NCLS  5
#define DHW   (96 * 160 * 160)          /* 2,457,600 voxels per batch      */
#define GROUPS_PER_BATCH (DHW / 4)      /* 614,400 groups of 4 voxels      */
#define TOTAL_GROUPS (BATCH * GROUPS_PER_BATCH)

// Workspace: ws[0..59]  = s_sum rows 0..9 (k + 5*b), cols 0..5 (col5 = nvox)
//            ws[60..119]= t_sum rows 0..9, cols 0..5 (col5 unused/0)
__global__ void kd_zero_ws(float* __restrict__ ws) {
    int i = threadIdx.x;
    if (i < 120) ws[i] = 0.0f;
}

__global__ __launch_bounds__(256) void kd_reduce(const float* __restrict__ src,
                                                 const float* __restrict__ tgt,
                                                 const int*   __restrict__ gt,
                                                 float*       __restrict__ ws)
{
    const int lane  = threadIdx.x & 31;
    const int half  = lane >> 4;        // 0: K=0,1   1: K=2,3
    const int m     = lane & 15;        // A-row (class slot) == B-col (channel)
    const int wave  = blockIdx.x * (blockDim.x >> 5) + (threadIdx.x >> 5);
    const int nwav  = gridDim.x * (blockDim.x >> 5);

    v8f acc_s = {};                     // 16x16 f32 accumulators
    v8f acc_t = {};

    const int  cplane   = (m < NCLS) ? m : 0;          // clamp OOB channel lanes
    const bool is_chan  = (m < NCLS);
    const float ones    = (m == NCLS) ? 1.0f : 0.0f;   // count column

    for (int g = wave; g < TOTAL_GROUPS; g += nwav) {  // wave-uniform loop
        const int  b   = (g >= GROUPS_PER_BATCH) ? 1 : 0;
        const long vox = (long)(g - b * GROUPS_PER_BATCH) * 4 + 2 * half;

        // gt for this lane's two K-slots (int32 on device)
        const int* gp = gt + (long)b * DHW + vox;
        const int  g0 = gp[0];
        const int  g1 = gp[1];

        // channel-plane values for this lane's two K-slots (8B aligned)
        const float* sp = src + ((long)b * NCLS + cplane) * (long)DHW + vox;
        const float* tp = tgt + ((long)b * NCLS + cplane) * (long)DHW + vox;
        const v2f sv = *(const v2f*)sp;
        const v2f tv = *(const v2f*)tp;

        // A: one-hot rows (class + 5*batch); B: channels (+ ones column for nvox)
        v2f a, bs, bt;
        const int r0 = g0 + NCLS * b;
        const int r1 = g1 + NCLS * b;
        a.x  = (m == r0) ? 1.0f : 0.0f;
        a.y  = (m == r1) ? 1.0f : 0.0f;
        bs.x = is_chan ? sv.x : ones;
        bs.y = is_chan ? sv.y : ones;
        bt.x = is_chan ? tv.x : 0.0f;
        bt.y = is_chan ? tv.y : 0.0f;

        // D = A*B + C  (full f32, EXEC all-ones here: loop cond is wave-uniform)
        acc_s = __builtin_amdgcn_wmma_f32_16x16x4_f32(false, a, false, bs,
                                                      (short)0, acc_s, false, false);
        acc_t = __builtin_amdgcn_wmma_f32_16x16x4_f32(false, a, false, bt,
                                                      (short)0, acc_t, false, false);
    }

    // C/D layout: VGPR r -> M = r + 8*half, N = lane&15. Keep M<10, N<6.
    if (m < 6) {
#pragma unroll
        for (int r = 0; r < 8; ++r) {
            const int M = r + 8 * half;
            if (M < 10) {
                atomicAdd(&ws[M * 6 + m],      acc_s[r]);
                atomicAdd(&ws[60 + M * 6 + m], acc_t[r]);
            }
        }
    }
}

__global__ void kd_epilogue(const float* __restrict__ ws, float* __restrict__ out)
{
    if (threadIdx.x != 0 || blockIdx.x != 0) return;
    const float invT = 0.5f;            // temperature = 2.0
    const float eps  = 1e-6f;
    float total = 0.0f;
    for (int row = 0; row < 10; ++row) {                 // row = k + 5*b
        const float denom = ws[row * 6 + 5] + eps;       // nvox + eps
        float s[4], t[4];
        for (int c = 0; c < 4; ++c) {                    // drop channel 0
            s[c] = (ws[row * 6 + (c + 1)] / denom) * invT;
            t[c] = (ws[60 + row * 6 + (c + 1)] / denom) * invT;
        }
        float ms = s[0], mt = t[0];
        for (int c = 1; c < 4; ++c) { ms = fmaxf(ms, s[c]); mt = fmaxf(mt, t[c]); }
        float es = 0.0f, et = 0.0f;
        for (int c = 0; c < 4; ++c) { es += expf(s[c] - ms); et += expf(t[c] - mt); }
        const float lse_s = logf(es) + ms;
        const float lse_t = logf(et) + mt;
        for (int c = 0; c < 4; ++c) {
            const float slp = s[c] - lse_s;
            const float tlp = t[c] - lse_t;
            total += expf(tlp) * (tlp - slp) + expf(slp) * (slp - tlp);
        }
    }
    // / batch(2) per KL, /2 for (kl_st+kl_ts)/2, /5 for mean over k  -> /20
    out[0] = total * 0.05f;
}

extern "C" void kernel_launch(void* const* d_in, const int* in_sizes, int n_in,
                              void* d_out, int out_size, void* d_ws, size_t ws_size,
                              hipStream_t stream)
{
    (void)in_sizes; (void)n_in; (void)out_size; (void)ws_size;
    const float* src = (const float*)d_in[0];
    const float* tgt = (const float*)d_in[1];
    const int*   gt  = (const int*)d_in[2];
    float*       ws  = (float*)d_ws;
    float*       out = (float*)d_out;

    kd_zero_ws<<<1, 128, 0, stream>>>(ws);
    kd_reduce<<<1024, 256, 0, stream>>>(src, tgt, gt, ws);
    kd_epilogue<<<1, 32, 0, stream>>>(ws, out);
}